// SpikingConv3DLayer_56023553409835
// MI455X (gfx1250) — compile-verified
//
#include <hip/hip_runtime.h>

typedef float v2f __attribute__((ext_vector_type(2)));
typedef float v8f __attribute__((ext_vector_type(8)));

#define EPSV 1e-8f

// Wave-private LDS fence: LDS is in-order per wave; just make sure outstanding
// DS ops complete and the compiler doesn't reorder memory ops across it.
__device__ __forceinline__ void wave_lds_fence() {
  asm volatile("s_wait_dscnt 0" ::: "memory");
}

// ---------------------------------------------------------------------------
// K0: Gram matrix d[a][f] = sum_i Wflat[a][i] * Wflat[f][i], Wflat = (32,27)
// ---------------------------------------------------------------------------
__global__ void gram_kernel(const float* __restrict__ w, float* __restrict__ dmat) {
  const int f = threadIdx.x;   // column
  const int a = threadIdx.y;   // row
  float s = 0.f;
#pragma unroll
  for (int i = 0; i < 27; ++i) s += w[a * 27 + i] * w[f * 27 + i];
  dmat[a * 32 + f] = s;
}

// ---------------------------------------------------------------------------
// Fused conv3x3 + leaky spiking scan. One wave per 16-pixel row segment.
//   B=8, Cin=3, T=32, H=W=64, Cout=32.
// ---------------------------------------------------------------------------
__global__ __launch_bounds__(128)
void snn_scan_kernel(const float* __restrict__ x, const float* __restrict__ w,
                     const float* __restrict__ betap, const float* __restrict__ bias,
                     const float* __restrict__ dmat, float* __restrict__ out) {
  constexpr int T = 32, H = 64, W = 64, Cin = 3, Cout = 32;
  constexpr int XROWSTR = 20;            // padded x-tile row stride (floats)
  constexpr int XBUF = 9 * XROWSTR;      // 180 floats
  constexpr int WAVE_LDS = 704;          // 180 (xbuf) + 512 (stage) + pad

  __shared__ float lds[4 * WAVE_LDS];

  const int lane = threadIdx.x & 31;
  const int wv   = threadIdx.x >> 5;
  const int gw   = blockIdx.x * 4 + wv;      // 0..2047
  const int b    = gw >> 8;                  // / 256
  const int rem  = gw & 255;
  const int h    = rem >> 2;                 // row 0..63
  const int w0   = (rem & 3) << 4;           // col base 0,16,32,48

  float* xbuf  = lds + wv * WAVE_LDS;        // x halo tile (9 x 20)
  float* stage = xbuf + XBUF;                // 32 x 16 matrix staging

  const int ln = lane & 15;
  const int hi = lane >> 4;

  // Conv weights for this lane's output channel (o = lane): 27 VGPRs.
  float wreg[27];
#pragma unroll
  for (int i = 0; i < 27; ++i) wreg[i] = w[lane * 27 + i];

  // A operand = d (symmetric Gram), WMMA 16x16x4 A layout:
  //   lane -> M = ln, VGPR j -> K = 4*kk + j + 2*hi
  v2f A0[8], A1[8];
#pragma unroll
  for (int kk = 0; kk < 8; ++kk) {
    const int c0 = 4 * kk + 2 * hi;
    A0[kk][0] = dmat[(c0 + 0) * 32 + ln];
    A0[kk][1] = dmat[(c0 + 1) * 32 + ln];
    A1[kk][0] = dmat[(c0 + 0) * 32 + 16 + ln];
    A1[kk][1] = dmat[(c0 + 1) * 32 + 16 + ln];
  }

  const float beta   = betap[0];
  const float ombeta = 1.f - beta;
  const float invn   = 1.f / (dmat[lane * 32 + lane] + EPSV);
  const float bl     = bias[lane];

  // Membrane state (channel = lane, 16 pixels) and zeroed spike staging.
  float mem[16];
#pragma unroll
  for (int p = 0; p < 16; ++p) mem[p] = 0.f;
  {
    const float4 z = make_float4(0.f, 0.f, 0.f, 0.f);
    float4* srow = reinterpret_cast<float4*>(stage + lane * 16);
#pragma unroll
    for (int q = 0; q < 4; ++q) srow[q] = z;
  }
  wave_lds_fence();

  const float* xb = x + (size_t)b * Cin * T * H * W;
  float* outb = out + (size_t)((b * Cout + lane) * T) * H * W + h * W + w0;

  for (int t = 0; t < T; ++t) {
    // ---- 1) x halo tile -> LDS (9 rows x 20 cols, zero padded) ----
#pragma unroll
    for (int i = 0; i < 6; ++i) {
      const int idx = lane + 32 * i;
      if (idx < 180) {
        const int row = idx / 20, col = idx % 20;   // row = ci*3 + kh
        const int ci = row / 3, kh = row % 3;
        const int hh = h + kh - 1;
        const int ww = w0 + col - 1;
        float v = 0.f;
        if (col < 18 && hh >= 0 && hh < H && ww >= 0 && ww < W)
          v = xb[((ci * T + t) * H + hh) * W + ww];
        xbuf[row * XROWSTR + col] = v;
      }
    }
    wave_lds_fence();

    // ---- 2) conv 3x3 (channel-per-lane, 16 pixels, 432 FMAs) ----
    float cv[16];
#pragma unroll
    for (int p = 0; p < 16; ++p) cv[p] = 0.f;
#pragma unroll
    for (int r9 = 0; r9 < 9; ++r9) {
      float xr[20];
      const float4* xrow = reinterpret_cast<const float4*>(xbuf + r9 * XROWSTR);
#pragma unroll
      for (int q = 0; q < 5; ++q) {
        const float4 tq = xrow[q];
        xr[4 * q + 0] = tq.x; xr[4 * q + 1] = tq.y;
        xr[4 * q + 2] = tq.z; xr[4 * q + 3] = tq.w;
      }
#pragma unroll
      for (int kw = 0; kw < 3; ++kw) {
        const float wvv = wreg[r9 * 3 + kw];
#pragma unroll
        for (int p = 0; p < 16; ++p) cv[p] = fmaf(xr[p + kw], wvv, cv[p]);
      }
    }

    // ---- 3) rst^T(32x16) = d(32x32) @ spk^T(32x16): 16x v_wmma_f32_16x16x4_f32
    //      B layout: lane -> N = ln, VGPR j -> K = 4*kk + j + 2*hi
    v8f r0 = {};
    v8f r1 = {};
#pragma unroll
    for (int kk = 0; kk < 8; ++kk) {
      const int c0 = 4 * kk + 2 * hi;
      v2f bv;
      bv[0] = stage[(c0 + 0) * 16 + ln];
      bv[1] = stage[(c0 + 1) * 16 + ln];
      r0 = __builtin_amdgcn_wmma_f32_16x16x4_f32(false, A0[kk], false, bv,
                                                 (short)0, r0, false, false);
      r1 = __builtin_amdgcn_wmma_f32_16x16x4_f32(false, A1[kk], false, bv,
                                                 (short)0, r1, false, false);
    }

    // ---- 4) stage rst: C/D layout -> row-major, reload per-channel row ----
    wave_lds_fence();
#pragma unroll
    for (int r = 0; r < 8; ++r) {
      stage[(r + 8 * hi     ) * 16 + ln] = r0[r];   // channels 0..15
      stage[(r + 8 * hi + 16) * 16 + ln] = r1[r];   // channels 16..31
    }
    wave_lds_fence();
    float rst[16];
    {
      const float4* srow = reinterpret_cast<const float4*>(stage + lane * 16);
#pragma unroll
      for (int q = 0; q < 4; ++q) {
        const float4 tq = srow[q];
        rst[4 * q + 0] = tq.x; rst[4 * q + 1] = tq.y;
        rst[4 * q + 2] = tq.z; rst[4 * q + 3] = tq.w;
      }
    }

    // ---- 5) membrane update, spike, write out, restage spk ----
    float4 so[4];
#pragma unroll
    for (int q = 0; q < 4; ++q) {
      float tmp[4];
#pragma unroll
      for (int j = 0; j < 4; ++j) {
        const int p = 4 * q + j;
        const float m = (mem[p] - rst[p]) * beta + cv[p] * ombeta;
        mem[p] = m;
        tmp[j] = ((m * invn - bl) > 0.f) ? 1.f : 0.f;
      }
      so[q] = make_float4(tmp[0], tmp[1], tmp[2], tmp[3]);
    }
    float4* orow = reinterpret_cast<float4*>(outb + t * H * W);
    float4* srow = reinterpret_cast<float4*>(stage + lane * 16);
#pragma unroll
    for (int q = 0; q < 4; ++q) {
      orow[q] = so[q];   // coalesced 64B/lane spike output
      srow[q] = so[q];   // spk^T staging for next step's B operand
    }
    wave_lds_fence();
  }
}

// ---------------------------------------------------------------------------
extern "C" void kernel_launch(void* const* d_in, const int* in_sizes, int n_in,
                              void* d_out, int out_size, void* d_ws, size_t ws_size,
                              hipStream_t stream) {
  (void)in_sizes; (void)n_in; (void)out_size; (void)ws_size;
  const float* x    = (const float*)d_in[0];   // (8,3,32,64,64)
  const float* w    = (const float*)d_in[1];   // (32,3,1,3,3)
  const float* beta = (const float*)d_in[2];   // (1,)
  const float* b    = (const float*)d_in[3];   // (32,)
  float* dmat = (float*)d_ws;                  // 32x32 Gram matrix (4 KB)
  float* out  = (float*)d_out;                 // (8,32,32,64,64)

  gram_kernel<<<1, dim3(32, 32), 0, stream>>>(w, dmat);
  snn_scan_kernel<<<512, 128, 0, stream>>>(x, w, beta, b, dmat, out);
}